// GCN_loop_43739946943353
// MI455X (gfx1250) — compile-verified
//
#include <hip/hip_runtime.h>
#include <stdint.h>

#define N_NODES   100000
#define N_EDGES   3200000
#define FEAT      128
#define N_GRAPHS  512
#define FILLW     2.0f

// LDS layout for the GEMM (dynamic LDS, carved manually)
#define A_ROW_DW   132                      // 128 + 4 pad dwords  -> banks (4l+k)%64, halves disjoint
#define A_BYTES    (128 * A_ROW_DW * 4)     // 67,584 B
#define W_PAIR_DW  288                      // 256 + 32 pad dwords -> 288%64==32, halves disjoint
#define W_BYTES    (64 * W_PAIR_DW * 4)     // 73,728 B
#define GEMM_LDS   (A_BYTES + W_BYTES)      // 141,312 B (<320KB WGP budget)

typedef __attribute__((ext_vector_type(2))) float v2f;
typedef __attribute__((ext_vector_type(8))) float v8f;

// ---------------------------------------------------------------- degree / dinv
__global__ __launch_bounds__(256) void k_init_deg(float* deg) {
  int i = blockIdx.x * 256 + threadIdx.x;
  if (i < N_NODES) deg[i] = FILLW;               // self-loop weight (improved=True)
}

__global__ __launch_bounds__(256) void k_edge_deg(const long long* __restrict__ ei,
                                                  float* deg) {
  int e = blockIdx.x * 256 + threadIdx.x;
  if (e < N_EDGES) {
    int c = (int)ei[(size_t)N_EDGES + e];        // col
    atomicAdd(&deg[c], 1.0f);
  }
}

__global__ __launch_bounds__(256) void k_dinv(float* deg) {
  int i = blockIdx.x * 256 + threadIdx.x;
  if (i < N_NODES) {
    float d = deg[i];
    deg[i] = (d > 0.0f) ? rsqrtf(d) : 0.0f;      // becomes dinv in place
  }
}

// ---------------------------------------------------------------- dense GEMM (WMMA f32)
// T[M,128] = A[M,128] @ W[128,128], fp32, V_WMMA_F32_16X16X4_F32.
// Block = 256 threads = 8 waves; wave w computes rows [blk*128+16w, +16) x all 128 cols.
// A tile staged via async global->LDS DMA with padded rows; W staged pair-interleaved
// so every A/B fragment is one bank-conflict-free ds_load_b64.
__global__ __launch_bounds__(256) void k_gemm_wmma(const float* __restrict__ A,
                                                   const float* __restrict__ W,
                                                   float* __restrict__ T, int M) {
  extern __shared__ char smem[];
  float* Al = (float*)smem;                      // [128][A_ROW_DW]
  float* Wp = (float*)(smem + A_BYTES);          // [64][W_PAIR_DW] pair-interleaved
  const int      tid      = threadIdx.x;
  const int      blkRow0  = blockIdx.x * 128;
  const uint32_t ldsAbase = (uint32_t)(uintptr_t)Al;

  // ---- stage A tile (128x128 f32) with async global->LDS B128 copies -------
  // per-lane LDS dst lets us insert the 4-dword row pad during the DMA itself
#pragma unroll
  for (int it = 0; it < 16; ++it) {
    int i   = it * 256 + tid;                    // 4096 float4 chunks
    int r   = i >> 5;                            // 0..127 tile row
    int c4  = i & 31;                            // 0..31 float4 within row
    int gr  = blkRow0 + r;
    if (gr > M - 1) gr = M - 1;                  // clamp tail block (rows never stored)
    uint64_t gaddr = (uint64_t)(uintptr_t)(A + (size_t)gr * FEAT + c4 * 4);
    uint32_t laddr = ldsAbase + (uint32_t)(r * (A_ROW_DW * 4) + c4 * 16);
    asm volatile("global_load_async_to_lds_b128 %0, %1, off"
                 :: "v"(laddr), "v"(gaddr) : "memory");
  }

  // ---- stage W pair-interleaved: Wp[p][2n+j] = W[2p+j][n] ------------------
#pragma unroll
  for (int it = 0; it < 8; ++it) {
    int i  = it * 256 + tid;                     // 2048 tasks
    int p  = i >> 5;                             // 0..63 k-pair
    int n4 = i & 31;                             // 0..31 col group of 4
    float4 a = *(const float4*)(W + (size_t)(2 * p)     * FEAT + n4 * 4);
    float4 b = *(const float4*)(W + (size_t)(2 * p + 1) * FEAT + n4 * 4);
    float* dst = Wp + (size_t)p * W_PAIR_DW + n4 * 8;
    *(float4*)(dst + 0) = make_float4(a.x, b.x, a.y, b.y);
    *(float4*)(dst + 4) = make_float4(a.z, b.z, a.w, b.w);
  }

  asm volatile("s_wait_asynccnt 0" ::: "memory");
  __syncthreads();

  const int wave  = tid >> 5;
  const int lane  = tid & 31;
  const int m16   = lane & 15;
  const int half  = lane >> 4;                   // 0: K=kk+0/1, 1: K=kk+2/3 (ISA 7.12.2)
  const int ncol  = lane & 15;

  const float* arow = Al + (size_t)(wave * 16 + m16) * A_ROW_DW + (half << 1);

  v8f acc[8] = {};                               // 8 N-tiles of 16 -> full 128 columns

  for (int kk = 0; kk < FEAT; kk += 4) {
    // A 16x4 fragment: one aligned ds_load_b64 per lane
    v2f a = *(const v2f*)(arow + kk);
    const float* bbase = Wp + (size_t)((kk >> 1) + half) * W_PAIR_DW + (ncol << 1);
#pragma unroll
    for (int nt = 0; nt < 8; ++nt) {
      // B 4x16 fragment: pair-interleaved -> one aligned ds_load_b64 per lane
      v2f b = *(const v2f*)(bbase + nt * 32);
      acc[nt] = __builtin_amdgcn_wmma_f32_16x16x4_f32(
          /*neg_a=*/false, a, /*neg_b=*/false, b,
          /*c_mod=*/(short)0, acc[nt], /*reuse_a=*/false, /*reuse_b=*/false);
    }
  }

  // D layout: VGPR j -> rows row0+j (lanes 0-15) / row0+j+8 (lanes 16-31)
  const int row0 = blkRow0 + wave * 16;
  const int srow = row0 + (half << 3);
#pragma unroll
  for (int nt = 0; nt < 8; ++nt) {
    const int n = nt * 16 + ncol;
#pragma unroll
    for (int j = 0; j < 8; ++j) {
      const int m = srow + j;
      if (m < M) T[(size_t)m * FEAT + n] = acc[nt][j];
    }
  }
}

// ---------------------------------------------------------------- aggregation
__global__ __launch_bounds__(256) void k_self_init(const float* __restrict__ t,
                                                   const float* __restrict__ dinv,
                                                   float* __restrict__ agg) {
  int idx = blockIdx.x * 256 + threadIdx.x;      // N*128 threads exactly
  int n = idx >> 7;
  float di = dinv[n];
  agg[idx] = t[idx] * (FILLW * di * di);         // self-loop contribution
}

__global__ __launch_bounds__(256) void k_edge_scatter(const float* __restrict__ t,
                                                      const long long* __restrict__ ei,
                                                      const float* __restrict__ dinv,
                                                      float* __restrict__ agg) {
  int gid  = blockIdx.x * 256 + threadIdx.x;     // E*32 threads exactly
  int e    = gid >> 5;
  int lane = gid & 31;
  int r = (int)ei[e];
  int c = (int)ei[(size_t)N_EDGES + e];
  float w = dinv[r] * dinv[c];
  float4 v = ((const float4*)(t + (size_t)r * FEAT))[lane];
  float* dst = agg + (size_t)c * FEAT + lane * 4;
  atomicAdd(dst + 0, v.x * w);
  atomicAdd(dst + 1, v.y * w);
  atomicAdd(dst + 2, v.z * w);
  atomicAdd(dst + 3, v.w * w);
}

__global__ __launch_bounds__(256) void k_bias_tanh(float* __restrict__ h,
                                                   const float* __restrict__ b) {
  int idx = blockIdx.x * 256 + threadIdx.x;      // N*128 threads exactly
  int f = idx & (FEAT - 1);
  h[idx] = tanhf(h[idx] + b[f]);
}

// ---------------------------------------------------------------- pooling
__global__ __launch_bounds__(256) void k_pool_init(unsigned* maxk, float* sums, int* cnt) {
  int idx = blockIdx.x * 256 + threadIdx.x;
  if (idx < N_GRAPHS * FEAT) { maxk[idx] = 0u; sums[idx] = 0.0f; }
  if (idx < N_GRAPHS) cnt[idx] = 0;
}

__device__ __forceinline__ unsigned enc_f32(float f) {
  unsigned u = __float_as_uint(f);
  return (u & 0x80000000u) ? ~u : (u | 0x80000000u);   // order-preserving
}
__device__ __forceinline__ float dec_f32(unsigned k) {
  return (k & 0x80000000u) ? __uint_as_float(k & 0x7fffffffu) : __uint_as_float(~k);
}

__global__ __launch_bounds__(256) void k_pool_scatter(const float* __restrict__ h,
                                                      const long long* __restrict__ batch,
                                                      unsigned* __restrict__ maxk,
                                                      float* __restrict__ sums,
                                                      int* __restrict__ cnt) {
  int idx = blockIdx.x * 256 + threadIdx.x;      // N*128 threads exactly
  int n = idx >> 7, f = idx & (FEAT - 1);
  int g = (int)batch[n];
  float v = h[idx];
  atomicMax(&maxk[g * FEAT + f], enc_f32(v));
  atomicAdd(&sums[g * FEAT + f], v);
  if (f == 0) atomicAdd(&cnt[g], 1);
}

__global__ __launch_bounds__(256) void k_final(const unsigned* __restrict__ maxk,
                                               const float* __restrict__ sums,
                                               const int* __restrict__ cnt,
                                               const float* __restrict__ Wout,
                                               const float* __restrict__ bout,
                                               float* __restrict__ out) {
  int g = blockIdx.x * 8 + (threadIdx.x >> 5);   // one wave32 per graph
  int lane = threadIdx.x & 31;
  if (g >= N_GRAPHS) return;
  float c = (float)cnt[g];
  float accv = 0.0f;
#pragma unroll
  for (int j0 = 0; j0 < FEAT; j0 += 32) {
    int j = j0 + lane;
    float mx = (c > 0.0f) ? dec_f32(maxk[g * FEAT + j]) : 0.0f;
    float mean = sums[g * FEAT + j] / fmaxf(c, 1.0f);
    accv += mx * Wout[j] + mean * Wout[FEAT + j];
  }
  for (int off = 16; off > 0; off >>= 1) accv += __shfl_xor(accv, off, 32);
  if (lane == 0) out[g] = accv + bout[0];
}

// ---------------------------------------------------------------- host launcher
extern "C" void kernel_launch(void* const* d_in, const int* in_sizes, int n_in,
                              void* d_out, int out_size, void* d_ws, size_t ws_size,
                              hipStream_t stream) {
  (void)in_sizes; (void)n_in; (void)out_size; (void)ws_size;
  const float*     x     = (const float*)d_in[0];
  const long long* ei    = (const long long*)d_in[1];
  const long long* batch = (const long long*)d_in[2];
  const float*     W0    = (const float*)d_in[3];
  const float*     b0    = (const float*)d_in[4];
  const float*     Wh    = (const float*)d_in[5];
  const float*     bh    = (const float*)d_in[6];
  const float*     Wout  = (const float*)d_in[7];
  const float*     bout  = (const float*)d_in[8];
  float*           out   = (float*)d_out;

  // workspace carve-up (256B aligned)
  char* p = (char*)d_ws;
  auto carve = [&p](size_t bytes) -> void* {
    void* r = (void*)p;
    p += (bytes + 255) & ~(size_t)255;
    return r;
  };
  float*    dinv = (float*)carve((size_t)N_NODES * 4);
  float*    t    = (float*)carve((size_t)N_NODES * FEAT * 4);
  float*    hA   = (float*)carve((size_t)N_NODES * FEAT * 4);
  float*    hB   = (float*)carve((size_t)N_NODES * FEAT * 4);
  unsigned* maxk = (unsigned*)carve((size_t)N_GRAPHS * FEAT * 4);
  float*    sums = (float*)carve((size_t)N_GRAPHS * FEAT * 4);
  int*      cnt  = (int*)carve((size_t)N_GRAPHS * 4);

  const int nodeBlks  = (N_NODES + 255) / 256;            // 391
  const int edgeBlks  = (N_EDGES + 255) / 256;            // 12500
  const int featBlks  = (N_NODES * FEAT) / 256;           // 50000
  const int scatBlks  = (N_EDGES / 256) * 32;             // 400000
  const int gemmBlks  = (N_NODES + 127) / 128;            // 782

  // 1) GCN normalization
  k_init_deg<<<nodeBlks, 256, 0, stream>>>(dinv);
  k_edge_deg<<<edgeBlks, 256, 0, stream>>>(ei, dinv);
  k_dinv<<<nodeBlks, 256, 0, stream>>>(dinv);

  // 2) three GCN layers
  const float* cur = x;
  float* hbuf[2] = { hA, hB };
  for (int l = 0; l < 3; ++l) {
    const float* Wl = (l == 0) ? W0 : Wh + (size_t)(l - 1) * FEAT * FEAT;
    const float* bl = (l == 0) ? b0 : bh + (size_t)(l - 1) * FEAT;
    float* agg = hbuf[l & 1];
    k_gemm_wmma   <<<gemmBlks, 256, GEMM_LDS, stream>>>(cur, Wl, t, N_NODES);
    k_self_init   <<<featBlks, 256, 0, stream>>>(t, dinv, agg);
    k_edge_scatter<<<scatBlks, 256, 0, stream>>>(t, ei, dinv, agg);
    k_bias_tanh   <<<featBlks, 256, 0, stream>>>(agg, bl);
    cur = agg;
  }

  // 3) graph pooling (max + mean) and readout
  k_pool_init   <<<(N_GRAPHS * FEAT) / 256, 256, 0, stream>>>(maxk, sums, cnt);
  k_pool_scatter<<<featBlks, 256, 0, stream>>>(cur, batch, maxk, sums, cnt);
  k_final       <<<N_GRAPHS / 8, 256, 0, stream>>>(maxk, sums, cnt, Wout, bout, out);
}